// FeaturesEmbedding_17746804867489
// MI455X (gfx1250) — compile-verified
//
#include <hip/hip_runtime.h>

typedef __attribute__((ext_vector_type(2))) float v2f;
typedef __attribute__((ext_vector_type(8))) float v8f;

#define BATCH     8192
#define NNZ       64
#define EMBED     64
#define FIELD_DIM 100000LL
#define NOUT_F    15

// One wave32 per batch element. Computes out[b] = A(16x64) x E(64x64) with
// V_WMMA_F32_16X16X4_F32, where A[f,n] = (x_field==f) ? x_val : 0 and E holds
// the gathered embedding rows. Rows 1..15 of the 16x64 product are stored.
__global__ __launch_bounds__(32)
void features_embedding_wmma(const long long* __restrict__ x_field,
                             const long long* __restrict__ x,
                             const float* __restrict__ x_val,
                             const float* __restrict__ table,
                             float* __restrict__ out)
{
    __shared__ int2      s_fv[NNZ];    // .x = field id, .y = bit-cast weight
    __shared__ long long s_row[NNZ];   // element offset of gathered row start

    const int b    = blockIdx.x;
    const int lane = threadIdx.x;

    // Stage indices/weights: 2 nnz entries per lane, coalesced.
    for (int n = lane; n < NNZ; n += 32) {
        const long long base = (long long)b * NNZ + n;
        const long long f = x_field[base];
        const long long i = x[base];
        const float     v = x_val[base];
        s_fv[n]  = make_int2((int)f, __float_as_int(v));
        s_row[n] = (i + f * FIELD_DIM) * EMBED;
    }
    __syncthreads();

    // Fire ALL 64 gather-row prefetches before any compute: the kernel is
    // latency-bound on independent table-row misses, so get every one in
    // flight up front (2 rows per lane).
    for (int n = lane; n < NNZ; n += 32) {
        __builtin_prefetch(table + s_row[n], 0, 3);
    }

    const int half = lane >> 4;   // lane group: rows K+0/1 vs K+2/3 of each 4-slice
    const int m    = lane & 15;   // A-matrix row (field id) / B,D column within tile

    v8f acc[4] = {};              // 4 tiles of 16 output columns (E = 64)

    for (int kk = 0; kk < NNZ / 4; ++kk) {
        const int n0 = kk * 4 + half * 2;   // nnz slot for A/B v-register 0
        const int n1 = n0 + 1;              // nnz slot for A/B v-register 1

        // A fragment (16x4 f32 layout): masked weights — folds the one-hot
        // field scatter and x_val weighting into the matmul. Branchless:
        // unconditional ds_load_b64 + v_cndmask + v_mul (no exec masking).
        const int2 fv0 = s_fv[n0];
        const int2 fv1 = s_fv[n1];
        v2f a;
        a.x = __int_as_float(fv0.y) * ((fv0.x == m) ? 1.0f : 0.0f);
        a.y = __int_as_float(fv1.y) * ((fv1.x == m) ? 1.0f : 0.0f);

        const float* r0 = table + s_row[n0];
        const float* r1 = table + s_row[n1];

        #pragma unroll
        for (int t = 0; t < 4; ++t) {
            const int col = t * 16 + m;
            v2f bm;                    // B fragment (4x16 f32): raw embedding data
            bm.x = r0[col];
            bm.y = r1[col];
            acc[t] = __builtin_amdgcn_wmma_f32_16x16x4_f32(
                /*neg_a=*/false, a, /*neg_b=*/false, bm,
                /*c_mod=*/(short)0, acc[t],
                /*reuse_a=*/false, /*reuse_b=*/false);
        }
    }

    // D layout: VGPR i of acc[t] holds output row M = i + 8*half, column t*16+m.
    // Row 0 (padding field) is dropped; rows 1..15 -> out[b, M-1, :].
    #pragma unroll
    for (int t = 0; t < 4; ++t) {
        #pragma unroll
        for (int i = 0; i < 8; ++i) {
            const int M = i + half * 8;
            if (M >= 1) {
                out[((long long)b * NOUT_F + (M - 1)) * EMBED + t * 16 + m] = acc[t][i];
            }
        }
    }
}

extern "C" void kernel_launch(void* const* d_in, const int* in_sizes, int n_in,
                              void* d_out, int out_size, void* d_ws, size_t ws_size,
                              hipStream_t stream) {
    const long long* x_field = (const long long*)d_in[0];
    const long long* x       = (const long long*)d_in[1];
    const float*     x_val   = (const float*)d_in[2];
    const float*     table   = (const float*)d_in[3];
    float*           out     = (float*)d_out;

    features_embedding_wmma<<<dim3(BATCH), dim3(32), 0, stream>>>(
        x_field, x, x_val, table, out);
}